// UnsupervisedRGCN_64407329571720
// MI455X (gfx1250) — compile-verified
//
#include <hip/hip_runtime.h>

// UnsupervisedRGCN on MI455X (gfx1250, wave32).
#define N_  20000
#define R_  4
#define S_  10
#define F_  128
#define E_  128
#define B_  4096
#define KREL (R_ * F_)      // 512
#define KTOT (KREL + F_)    // 640 : [agg(r=0..3) | self]
#define LDSTRIDE 650        // 650 % 64 = 10 -> conflict-free LDS row stride

typedef __attribute__((ext_vector_type(2))) float v2f;
typedef __attribute__((ext_vector_type(8))) float v8f;

// Software-pipelined fp32 WMMA span: C += A_lds[16 x 4k] * W[4k x 16].
//  Ap: lane's A pointer (&A[arow*LDSTRIDE + khalf]), +4 floats per k-step.
//  Wp: lane's B pointer (&W[khalf*E_ + col]),        +4*E_ floats per k-step.
// Register double-buffer: fragment k+1 is in flight while WMMA consumes k.
__device__ __forceinline__ void gemm_span(const float* __restrict__ Ap,
                                          const float* __restrict__ Wp,
                                          int ksteps, v8f& c)
{
    v2f a, b;
    a.x = Ap[0];  a.y = Ap[1];
    b.x = Wp[0];  b.y = Wp[E_];
    #pragma unroll 4
    for (int k = 1; k < ksteps; ++k) {
        v2f an, bn;
        an.x = Ap[4 * k];
        an.y = Ap[4 * k + 1];
        bn.x = Wp[(4 * k) * E_];
        bn.y = Wp[(4 * k + 1) * E_];
        c = __builtin_amdgcn_wmma_f32_16x16x4_f32(
                false, a, false, b, (short)0, c, false, false);
        a = an; b = bn;
    }
    c = __builtin_amdgcn_wmma_f32_16x16x4_f32(
            false, a, false, b, (short)0, c, false, false);
}

// Fused RGCN layer:
//   out[g,:] = relu( sum_r mean_s feat[neigh(g,r,s)] @ w_rel[r] + feat[node] @ w_self )
// node = nodelist ? nodelist[g] : g. Block = 256 threads handles 16 output rows.
__global__ __launch_bounds__(256)
void rgcn_layer_kernel(const float* __restrict__ feat,       // [*, F_]
                       const float* __restrict__ w_rel,      // [KREL, E_]
                       const float* __restrict__ w_self,     // [F_,  E_]
                       const int*   __restrict__ samp_neigh, // [N_, R_, S_]
                       const int*   __restrict__ nodelist,   // [rows] or nullptr
                       float*       __restrict__ out)        // [rows, E_]
{
    __shared__ float A[16 * LDSTRIDE];   // 16 x 640 fp32 A-tile (padded)

    const int tid = threadIdx.x;
    const int row = tid >> 4;            // 0..15 node row within tile
    const int l16 = tid & 15;
    const int g   = blockIdx.x * 16 + row;
    const int node = nodelist ? nodelist[g] : g;

    // ---- Phase 1: gather + mean neighbors per relation, plus self row ----
    const int f0 = l16 * 8;              // 8 floats per thread per row
    const int nbase = node * (R_ * S_);
    #pragma unroll
    for (int r = 0; r < R_; ++r) {
        float acc0 = 0.f, acc1 = 0.f, acc2 = 0.f, acc3 = 0.f;
        float acc4 = 0.f, acc5 = 0.f, acc6 = 0.f, acc7 = 0.f;
        #pragma unroll
        for (int s = 0; s < S_; ++s) {
            const int nb = samp_neigh[nbase + r * S_ + s];
            const float4* p = (const float4*)(feat + (long)nb * F_ + f0);
            float4 u = p[0];
            float4 v = p[1];
            acc0 += u.x; acc1 += u.y; acc2 += u.z; acc3 += u.w;
            acc4 += v.x; acc5 += v.y; acc6 += v.z; acc7 += v.w;
        }
        float* dst = &A[row * LDSTRIDE + r * F_ + f0];
        dst[0] = acc0 * 0.1f; dst[1] = acc1 * 0.1f;
        dst[2] = acc2 * 0.1f; dst[3] = acc3 * 0.1f;
        dst[4] = acc4 * 0.1f; dst[5] = acc5 * 0.1f;
        dst[6] = acc6 * 0.1f; dst[7] = acc7 * 0.1f;
    }
    {   // self features -> K = 512..639
        const float4* p = (const float4*)(feat + (long)node * F_ + f0);
        float4 u = p[0];
        float4 v = p[1];
        float* dst = &A[row * LDSTRIDE + KREL + f0];
        dst[0] = u.x; dst[1] = u.y; dst[2] = u.z; dst[3] = u.w;
        dst[4] = v.x; dst[5] = v.y; dst[6] = v.z; dst[7] = v.w;
    }
    __syncthreads();

    // ---- Phase 2: per-wave 16x16 fp32 WMMA tile over K = 640 ----
    const int wave    = tid >> 5;          // 0..7 -> 16-col output slab
    const int lane    = tid & 31;
    const int colbase = wave * 16;
    const int arow    = lane & 15;         // A row (M) owned by this lane
    const int khalf   = (lane >> 4) << 1;  // lanes 0-15: K+0/1 ; lanes 16-31: K+2/3
    const int col     = colbase + (lane & 15);

    v8f c = {};
    // Span 1: relation weights, K = 0..511 (128 steps), branch-free.
    gemm_span(&A[arow * LDSTRIDE + khalf],
              w_rel + (long)khalf * E_ + col,
              KREL / 4, c);
    // Span 2: self-loop weights, K = 512..639 (32 steps), branch-free.
    gemm_span(&A[arow * LDSTRIDE + KREL + khalf],
              w_self + (long)khalf * E_ + col,
              F_ / 4, c);

    // ---- ReLU + store. C layout: lanes 0-15 -> M = v, lanes 16-31 -> M = v+8 ----
    const int moff = (lane >> 4) * 8;
    const int ncol = colbase + (lane & 15);
    #pragma unroll
    for (int v = 0; v < 8; ++v) {
        const int m    = v + moff;
        const int grow = blockIdx.x * 16 + m;
        float val = c[v];
        val = val > 0.f ? val : 0.f;
        out[(long)grow * E_ + ncol] = val;
    }
}

// DistMult decoder: score[b] = sum_e sub[b,e] * rel_emb[rels[b],e] * obj[b,e]
__global__ __launch_bounds__(256)
void distmult_kernel(const float* __restrict__ sub,
                     const float* __restrict__ obj,
                     const float* __restrict__ rel_emb,
                     const int*   __restrict__ rels,
                     float*       __restrict__ out)
{
    const int wave = threadIdx.x >> 5;
    const int lane = threadIdx.x & 31;
    const int b = blockIdx.x * 8 + wave;
    if (b >= B_) return;
    const float* re = rel_emb + (long)rels[b] * E_;
    const float* sb = sub + (long)b * E_;
    const float* ob = obj + (long)b * E_;
    float s = 0.f;
    #pragma unroll
    for (int e0 = 0; e0 < E_; e0 += 32) {
        const int e = e0 + lane;
        s += sb[e] * re[e] * ob[e];
    }
    #pragma unroll
    for (int off = 16; off > 0; off >>= 1)
        s += __shfl_xor(s, off, 32);
    if (lane == 0) out[b] = s;
}

extern "C" void kernel_launch(void* const* d_in, const int* in_sizes, int n_in,
                              void* d_out, int out_size, void* d_ws, size_t ws_size,
                              hipStream_t stream) {
    const float* node_features = (const float*)d_in[0]; // [N, F]
    const float* w1            = (const float*)d_in[1]; // [R, F, E] == [KREL, E]
    const float* w1_self       = (const float*)d_in[2]; // [F, E]
    const float* w2            = (const float*)d_in[3]; // [R, E, E] == [KREL, E]
    const float* w2_self       = (const float*)d_in[4]; // [E, E]
    const float* rel_emb       = (const float*)d_in[5]; // [R, E]
    const int*   samp_neigh    = (const int*)  d_in[6]; // [N, R, S]
    const int*   sub_nodes     = (const int*)  d_in[7]; // [B]
    const int*   obj_nodes     = (const int*)  d_in[8]; // [B]
    const int*   rels          = (const int*)  d_in[9]; // [B]

    // Workspace carve: h1 [N,E] | sub_emb [B,E] | obj_emb [B,E]
    float* h1      = (float*)d_ws;
    float* sub_emb = h1 + (size_t)N_ * E_;
    float* obj_emb = sub_emb + (size_t)B_ * E_;

    // Layer 1: all nodes (20000 / 16 = 1250 tiles)
    rgcn_layer_kernel<<<N_ / 16, 256, 0, stream>>>(
        node_features, w1, w1_self, samp_neigh, nullptr, h1);

    // Layer 2: subject and object query nodes (4096 / 16 = 256 tiles each)
    rgcn_layer_kernel<<<B_ / 16, 256, 0, stream>>>(
        h1, w2, w2_self, samp_neigh, sub_nodes, sub_emb);
    rgcn_layer_kernel<<<B_ / 16, 256, 0, stream>>>(
        h1, w2, w2_self, samp_neigh, obj_nodes, obj_emb);

    // DistMult decoder (8 rows per 256-thread block)
    distmult_kernel<<<B_ / 8, 256, 0, stream>>>(
        sub_emb, obj_emb, rel_emb, rels, (float*)d_out);
}